// GATv2_Salt_34978213658794
// MI455X (gfx1250) — compile-verified
//
#include <hip/hip_runtime.h>
#include <math.h>
#include <stdint.h>

#define NN   50000
#define EE   800000
#define BB   500
#define HH   4
#define DD   32
#define CH   128     // H*D (all GEMMs here have 128 output cols)
#define FIN  64
#define HIDN 128
#define MAXK 128

typedef float    v2f __attribute__((ext_vector_type(2)));
typedef float    v8f __attribute__((ext_vector_type(8)));
typedef unsigned v4u __attribute__((ext_vector_type(4)));
typedef int      v4i __attribute__((ext_vector_type(4)));
typedef int      v8i __attribute__((ext_vector_type(8)));

// ---------- monotonic float <-> uint key (for atomic float max) ----------
__device__ __forceinline__ unsigned fkey(float f) {
    unsigned u = __float_as_uint(f);
    return (u & 0x80000000u) ? ~u : (u | 0x80000000u);
}
__device__ __forceinline__ float kinv(unsigned k) {
    return __uint_as_float((k & 0x80000000u) ? (k & 0x7fffffffu) : ~k);
}

// ---------- fills ----------
__global__ void fill_f32(float* __restrict__ p, float v, int n) {
    int i = blockIdx.x * blockDim.x + threadIdx.x;
    if (i < n) p[i] = v;
}
__global__ void fill_u32(unsigned* __restrict__ p, unsigned v, int n) {
    int i = blockIdx.x * blockDim.x + threadIdx.x;
    if (i < n) p[i] = v;
}

// ---------- fp32 WMMA GEMM: D[M,128] = A[M,K] @ B[K,128], row-major ----------
// Block = 256 threads (8 waves). Whole B panel staged into LDS via the Tensor
// Data Mover (TDM), then each wave computes one 16-row strip across all 128
// columns (8 accumulators), reusing each A fragment for 8 WMMAs.
__global__ void gemm_wmma_f32(const float* __restrict__ A, const float* __restrict__ Bm,
                              float* __restrict__ Dm, int M, int K, int tilesM) {
    __shared__ float Bs[MAXK * CH];   // 64 KB (<= 320 KB/WGP)

#if __has_builtin(__builtin_amdgcn_tensor_load_to_lds)
    if ((threadIdx.x >> 5) == 0) {    // wave 0 issues the DMA (EXEC-independent)
        unsigned      ldsOff = (unsigned)(uintptr_t)(void*)Bs;   // LDS aperture: addr[31:0]
        unsigned long ga     = (unsigned long)(uintptr_t)Bm;
        // ---- D# group 0: count=1(valid), lds_addr, global_addr[56:0], type=2 ----
        v4u g0;
        g0.x = 1u;
        g0.y = ldsOff;
        g0.z = (unsigned)(ga & 0xffffffffu);
        g0.w = (unsigned)((ga >> 32) & 0x01ffffffu) | (2u << 30);
        // ---- D# group 1: data_size=4B, tensor 128 x K, tile 128 x K, stride 128 ----
        v8i g1;
        g1[0] = (int)(2u << 16);                 // data_size = 2 -> 4 bytes
        g1[1] = (int)(128u << 16);               // tensor_dim0[15:0] = 128
        g1[2] = (int)(((unsigned)K) << 16);      // tensor_dim0 hi=0 | tensor_dim1[15:0]=K
        g1[3] = (int)(128u << 16);               // tensor_dim1 hi=0 | tile_dim0=128
        g1[4] = (int)(unsigned)K;                // tile_dim1 = K | tile_dim2 = 0
        g1[5] = 128;                             // tensor_dim0_stride[31:0] = 128
        g1[6] = 0;                               // stride hi | tensor_dim1_stride lo
        g1[7] = 0;
        v4i g2 = {0, 0, 0, 0};
        v4i g3 = {0, 0, 0, 0};
#if __clang_major__ >= 23
        v8i g4 = {0, 0, 0, 0, 0, 0, 0, 0};
        __builtin_amdgcn_tensor_load_to_lds(g0, g1, g2, g3, g4, 0);
#else
        __builtin_amdgcn_tensor_load_to_lds(g0, g1, g2, g3, 0);
#endif
        __builtin_amdgcn_s_wait_tensorcnt(0);
    }
#else
    for (int i = threadIdx.x; i < (K * CH) / 4; i += blockDim.x)
        ((float4*)Bs)[i] = ((const float4*)Bm)[i];
#endif
    __syncthreads();

    int wave  = threadIdx.x >> 5;
    int mTile = blockIdx.x * 8 + wave;            // wave-uniform
    if (mTile >= tilesM) return;                  // whole-wave exit, EXEC stays all-1
    int lane = threadIdx.x & 31;
    int half = lane >> 4;                         // 0: K pair (k,k+1)  1: (k+2,k+3)
    int l    = lane & 15;
    int row  = mTile * 16 + l; if (row > M - 1) row = M - 1;   // clamped loads
    const float* arow = A + (size_t)row * K;

    v8f acc[8];
#pragma unroll
    for (int nt = 0; nt < 8; ++nt) acc[nt] = (v8f){0.f,0.f,0.f,0.f,0.f,0.f,0.f,0.f};

    for (int k0 = 0; k0 < K; k0 += 4) {
        int ka = k0 + 2 * half;
        v2f a;
        a.x = arow[ka];
        a.y = arow[ka + 1];
#pragma unroll
        for (int nt = 0; nt < 8; ++nt) {
            v2f b;
            b.x = Bs[ka * CH + nt * 16 + l];
            b.y = Bs[(ka + 1) * CH + nt * 16 + l];
            acc[nt] = __builtin_amdgcn_wmma_f32_16x16x4_f32(false, a, false, b,
                                                            (short)0, acc[nt], false, false);
        }
    }

    int rbase = mTile * 16 + half * 8;
#pragma unroll
    for (int nt = 0; nt < 8; ++nt) {
        int col = nt * 16 + l;
#pragma unroll
        for (int m = 0; m < 8; ++m) {
            int r = rbase + m;
            if (r < M) Dm[(size_t)r * CH + col] = acc[nt][m];
        }
    }
}

// ---------- edge pass A: raw scores + global max ----------
__global__ void edge_score(const float* __restrict__ proj, const float* __restrict__ attn,
                           const int* __restrict__ src, const int* __restrict__ dst,
                           float* __restrict__ scores, unsigned* __restrict__ smaxk) {
    int t = blockIdx.x * blockDim.x + threadIdx.x;
    float s = -INFINITY;
    if (t < EE * HH) {
        int e = t >> 2, h = t & 3;
        int si = src[e], di = dst[e];
        const float4* ps = (const float4*)(proj + (size_t)si * CH + h * DD);
        const float4* pd = (const float4*)(proj + (size_t)di * CH + h * DD);
        const float4* pa = (const float4*)(attn + h * DD);
        float acc = 0.f;
#pragma unroll
        for (int i = 0; i < 8; ++i) {
            float4 a = ps[i], b = pd[i], w = pa[i];
            float v0 = a.x + b.x; v0 = v0 >= 0.f ? v0 : 0.2f * v0;
            float v1 = a.y + b.y; v1 = v1 >= 0.f ? v1 : 0.2f * v1;
            float v2 = a.z + b.z; v2 = v2 >= 0.f ? v2 : 0.2f * v2;
            float v3 = a.w + b.w; v3 = v3 >= 0.f ? v3 : 0.2f * v3;
            acc += w.x * v0 + w.y * v1 + w.z * v2 + w.w * v3;
        }
        scores[t] = acc;
        s = acc;
    }
#pragma unroll
    for (int m = 16; m; m >>= 1) {
        float o = __shfl_xor(s, m, 32);
        s = s > o ? s : o;
    }
    if ((threadIdx.x & 31) == 0) atomicMax(smaxk, fkey(s));
}

// ---------- edge pass B: exp + denominator scatter ----------
__global__ void edge_exp(float* __restrict__ scores, const unsigned* __restrict__ smaxk,
                         const int* __restrict__ dst, float* __restrict__ denom) {
    int t = blockIdx.x * blockDim.x + threadIdx.x;
    if (t >= EE * HH) return;
    float mx = kinv(*smaxk);
    int e = t >> 2, h = t & 3;
    float w = __expf(scores[t] - mx);
    scores[t] = w;
    unsafeAtomicAdd(&denom[(size_t)dst[e] * HH + h], w);
}

// ---------- edge pass C: alpha-weighted aggregation ----------
__global__ void edge_aggr(const float* __restrict__ scores, const float* __restrict__ denom,
                          const float* __restrict__ proj,
                          const int* __restrict__ src, const int* __restrict__ dst,
                          float* __restrict__ outacc) {
    int t = blockIdx.x * blockDim.x + threadIdx.x;
    if (t >= EE * HH) return;
    int e = t >> 2, h = t & 3;
    int di = dst[e];
    float alpha = scores[t] / (denom[(size_t)di * HH + h] + 1e-16f);
    const float4* ps = (const float4*)(proj + (size_t)src[e] * CH + h * DD);
    float* po = outacc + (size_t)di * CH + h * DD;
#pragma unroll
    for (int i = 0; i < 8; ++i) {
        float4 p = ps[i];
        unsafeAtomicAdd(po + 4 * i + 0, p.x * alpha);
        unsafeAtomicAdd(po + 4 * i + 1, p.y * alpha);
        unsafeAtomicAdd(po + 4 * i + 2, p.z * alpha);
        unsafeAtomicAdd(po + 4 * i + 3, p.w * alpha);
    }
}

// ---------- epilogue: concat layers (0,1): h = prelu(out + res + bias) ----------
__global__ void epi_concat(const float* __restrict__ outacc, const float* __restrict__ res,
                           const float* __restrict__ bias, const float* __restrict__ pr,
                           float* __restrict__ hout) {
    int i = blockIdx.x * blockDim.x + threadIdx.x;
    if (i >= NN * CH) return;
    int c = i & (CH - 1);
    float a = pr[0];
    float v = outacc[i] + res[i] + bias[c];
    hout[i] = v >= 0.f ? v : a * v;
}

// ---------- epilogue layer 2: head mean, bias(D), prelu ----------
__global__ void epi_mean(const float* __restrict__ outacc, const float* __restrict__ hprev,
                         const float* __restrict__ bias, const float* __restrict__ pr,
                         float* __restrict__ h3) {
    int i = blockIdx.x * blockDim.x + threadIdx.x;
    if (i >= NN * DD) return;
    int n = i >> 5, d = i & 31;
    const float* po = outacc + (size_t)n * CH + d;
    const float* ph = hprev  + (size_t)n * CH + d;
    float s = 0.f;
#pragma unroll
    for (int h = 0; h < HH; ++h) s += po[h * DD] + ph[h * DD];
    float v = 0.25f * s + bias[d];
    float a = pr[0];
    h3[i] = v >= 0.f ? v : a * v;
}

// ---------- pooling: one wave per node, lane = feature dim ----------
__global__ void pool_kernel(const float* __restrict__ h3, const float* __restrict__ awW,
                            const float* __restrict__ awb,
                            const int* __restrict__ bidx, const int* __restrict__ ncomp,
                            unsigned* __restrict__ pmaxk, float* __restrict__ psum) {
    int n = blockIdx.x * (blockDim.x >> 5) + (threadIdx.x >> 5);
    if (n >= NN) return;                  // wave-uniform
    int d = threadIdx.x & 31;
    float v = h3[(size_t)n * DD + d];
    float p = v * awW[d];
#pragma unroll
    for (int m = 16; m; m >>= 1) p += __shfl_xor(p, m, 32);
    float w = 1.f / (1.f + __expf(-(p + awb[0])));
    int seg = bidx[n] + ncomp[n] * BB;
    atomicMax(&pmaxk[(size_t)seg * DD + d], fkey(v));
    unsafeAtomicAdd(&psum[(size_t)seg * DD + d], w * v);
}

// ---------- assemble g[B, 4D] = [pmax[b], psum[b], pmax[b+B], psum[b+B]] ----------
__global__ void build_g(const unsigned* __restrict__ pmaxk, const float* __restrict__ psum,
                        float* __restrict__ g) {
    int i = blockIdx.x * blockDim.x + threadIdx.x;
    if (i >= BB * 4 * DD) return;
    int b = i >> 7, c = i & 127;
    float v;
    if (c < 32)       v = kinv(pmaxk[(size_t)b * DD + c]);
    else if (c < 64)  v = psum[(size_t)b * DD + (c - 32)];
    else if (c < 96)  v = kinv(pmaxk[(size_t)(b + BB) * DD + (c - 64)]);
    else              v = psum[(size_t)(b + BB) * DD + (c - 96)];
    g[i] = v;
}

// ---------- in-place bias + prelu ----------
__global__ void bias_prelu(float* __restrict__ buf, const float* __restrict__ bias,
                           const float* __restrict__ pr, int total, int colsMask) {
    int i = blockIdx.x * blockDim.x + threadIdx.x;
    if (i >= total) return;
    float a = pr[0];
    float v = buf[i] + bias[i & colsMask];
    buf[i] = v >= 0.f ? v : a * v;
}

// ---------- final: out[b] = dot(hmid[b], W1) + b1 ----------
__global__ void final_dot(const float* __restrict__ hmid, const float* __restrict__ W1,
                          const float* __restrict__ b1, float* __restrict__ out) {
    int b = blockIdx.x * (blockDim.x >> 5) + (threadIdx.x >> 5);
    if (b >= BB) return;
    int lane = threadIdx.x & 31;
    const float* row = hmid + (size_t)b * HIDN;
    float s = 0.f;
#pragma unroll
    for (int c = 0; c < HIDN; c += 32) s += row[c + lane] * W1[c + lane];
#pragma unroll
    for (int m = 16; m; m >>= 1) s += __shfl_xor(s, m, 32);
    if (lane == 0) out[b] = s + b1[0];
}

extern "C" void kernel_launch(void* const* d_in, const int* in_sizes, int n_in,
                              void* d_out, int out_size, void* d_ws, size_t ws_size,
                              hipStream_t stream) {
    const float* x     = (const float*)d_in[0];
    const float* W0    = (const float*)d_in[1];
    const float* resW0 = (const float*)d_in[2];
    const float* attn0 = (const float*)d_in[3];
    const float* b0    = (const float*)d_in[4];
    const float* pr0   = (const float*)d_in[5];
    const float* W1    = (const float*)d_in[6];
    const float* attn1 = (const float*)d_in[7];
    const float* b1    = (const float*)d_in[8];
    const float* pr1   = (const float*)d_in[9];
    const float* W2    = (const float*)d_in[10];
    const float* attn2 = (const float*)d_in[11];
    const float* b2    = (const float*)d_in[12];
    const float* pr2   = (const float*)d_in[13];
    const float* awW   = (const float*)d_in[14];
    const float* awb   = (const float*)d_in[15];
    const float* mlpW0 = (const float*)d_in[16];
    const float* mlpb0 = (const float*)d_in[17];
    const float* mlppr = (const float*)d_in[18];
    const float* mlpW1 = (const float*)d_in[19];
    const float* mlpb1 = (const float*)d_in[20];
    const int* esrc  = (const int*)d_in[21];
    const int* edst  = (const int*)d_in[22];
    const int* bidx  = (const int*)d_in[23];
    const int* ncomp = (const int*)d_in[24];
    float* out = (float*)d_out;

    // ---- carve workspace (256B aligned) ----
    char* ws = (char*)d_ws;
    size_t off = 0;
    auto alloc = [&](size_t bytes) -> void* {
        off = (off + 255) & ~(size_t)255;
        void* p = ws + off;
        off += bytes;
        return p;
    };
    float*    proj   = (float*)   alloc((size_t)NN * CH * 4);
    float*    outacc = (float*)   alloc((size_t)NN * CH * 4);
    float*    hA     = (float*)   alloc((size_t)NN * CH * 4);
    float*    hB     = (float*)   alloc((size_t)NN * CH * 4);   // doubles as layer-0 residual buf
    float*    scores = (float*)   alloc((size_t)EE * HH * 4);
    float*    denom  = (float*)   alloc((size_t)NN * HH * 4);
    float*    h3     = (float*)   alloc((size_t)NN * DD * 4);
    unsigned* pmaxk  = (unsigned*)alloc((size_t)2 * BB * DD * 4);
    float*    psum   = (float*)   alloc((size_t)2 * BB * DD * 4);
    float*    g      = (float*)   alloc((size_t)BB * 4 * DD * 4);
    float*    hmid   = (float*)   alloc((size_t)BB * HIDN * 4);
    unsigned* smaxk  = (unsigned*)alloc(256);

    const unsigned NEGINF_KEY = 0x007FFFFFu;  // fkey(-inf)
    const int TB = 256;
    auto blocks = [](long long n, int tb) { return (int)((n + tb - 1) / tb); };

    auto gemm = [&](const float* A, const float* Bm, float* Dm, int M, int K) {
        int tilesM = (M + 15) / 16;
        int nblk = (tilesM + 7) / 8;                     // 8 waves / block, 1 M-tile each
        gemm_wmma_f32<<<nblk, 256, 0, stream>>>(A, Bm, Dm, M, K, tilesM);
    };

    auto edge_layer = [&](const float* attn) {
        fill_u32<<<1, 32, 0, stream>>>(smaxk, NEGINF_KEY, 1);
        fill_f32<<<blocks((long long)NN * HH, TB), TB, 0, stream>>>(denom, 0.f, NN * HH);
        fill_f32<<<blocks((long long)NN * CH, TB), TB, 0, stream>>>(outacc, 0.f, NN * CH);
        int eb = blocks((long long)EE * HH, TB);
        edge_score<<<eb, TB, 0, stream>>>(proj, attn, esrc, edst, scores, smaxk);
        edge_exp  <<<eb, TB, 0, stream>>>(scores, smaxk, edst, denom);
        edge_aggr <<<eb, TB, 0, stream>>>(scores, denom, proj, esrc, edst, outacc);
    };

    // ---- layer 0: proj = x@W0, res = x@res_W0 ----
    gemm(x, W0, proj, NN, FIN);
    gemm(x, resW0, hB /*res*/, NN, FIN);
    edge_layer(attn0);
    epi_concat<<<blocks((long long)NN * CH, TB), TB, 0, stream>>>(outacc, hB, b0, pr0, hA);

    // ---- layer 1 ----
    gemm(hA, W1, proj, NN, CH);
    edge_layer(attn1);
    epi_concat<<<blocks((long long)NN * CH, TB), TB, 0, stream>>>(outacc, hA, b1, pr1, hB);

    // ---- layer 2 (mean over heads) ----
    gemm(hB, W2, proj, NN, CH);
    edge_layer(attn2);
    epi_mean<<<blocks((long long)NN * DD, TB), TB, 0, stream>>>(outacc, hB, b2, pr2, h3);

    // ---- component-aware pooling ----
    fill_u32<<<blocks(2LL * BB * DD, TB), TB, 0, stream>>>(pmaxk, NEGINF_KEY, 2 * BB * DD);
    fill_f32<<<blocks(2LL * BB * DD, TB), TB, 0, stream>>>(psum, 0.f, 2 * BB * DD);
    pool_kernel<<<blocks((long long)NN, 8), TB, 0, stream>>>(h3, awW, awb, bidx, ncomp, pmaxk, psum);
    build_g<<<blocks((long long)BB * 4 * DD, TB), TB, 0, stream>>>(pmaxk, psum, g);

    // ---- prediction MLP ----
    gemm(g, mlpW0, hmid, BB, 4 * DD);
    bias_prelu<<<blocks((long long)BB * HIDN, TB), TB, 0, stream>>>(hmid, mlpb0, mlppr,
                                                                   BB * HIDN, HIDN - 1);
    final_dot<<<blocks((long long)BB, 8), TB, 0, stream>>>(hmid, mlpW1, mlpb1, out);
}